// MultiRetrievalAugmentedEmbeddingV2_25555055411289
// MI455X (gfx1250) — compile-verified
//
#include <hip/hip_runtime.h>
#include <cstdint>

typedef __attribute__((ext_vector_type(2))) float v2f;
typedef __attribute__((ext_vector_type(8))) float v8f;

#define B_    8
#define N_    50000
#define V_    768
#define A_    384
#define O_    384
#define TOPK_ 1000

// workspace layout (float offsets)
#define WS_Q1  0                         // 16 x 768 padded normalized v
#define WS_Q2  (WS_Q1 + 16 * V_)         // 16 x 384 padded normalized aud
#define WS_E1  (WS_Q2 + 16 * A_)         // 8 x N  exp(sim1) -> later mixed att
#define WS_E2  (WS_E1 + B_ * N_)         // 8 x N  exp(sim2)
#define WS_Z   (WS_E2 + B_ * N_)         // 16 (Z1[0..7], Z2[8..15])
#define WS_THR (WS_Z + 16)               // 8
#define WS_OIA (WS_THR + 8)              // 8 x 384

// ---------------------------------------------------------------- prep
__global__ void prep_kernel(const float* __restrict__ v,
                            const float* __restrict__ aud,
                            float* __restrict__ ws) {
  __shared__ float red[256];
  const int blk = blockIdx.x;
  const int t = threadIdx.x;
  if (blk < 16) {
    const float* src; float* dst; int K;
    if (blk < 8) { src = v + blk * V_;        dst = ws + WS_Q1 + blk * V_;        K = V_; }
    else         { src = aud + (blk - 8) * A_; dst = ws + WS_Q2 + (blk - 8) * A_; K = A_; }
    float s = 0.f;
    for (int i = t; i < K; i += 256) { float x = src[i]; s += x * x; }
    red[t] = s; __syncthreads();
    for (int off = 128; off > 0; off >>= 1) {
      if (t < off) red[t] += red[t + off];
      __syncthreads();
    }
    const float inv = 1.0f / fmaxf(sqrtf(red[0]), 1e-12f);
    for (int i = t; i < K; i += 256) dst[i] = src[i] * inv;
  } else {
    // zero pad rows 8..15 of both A panels + accumulators
    for (int i = t; i < 8 * V_;  i += 256) ws[WS_Q1 + 8 * V_ + i] = 0.f;
    for (int i = t; i < 8 * A_;  i += 256) ws[WS_Q2 + 8 * A_ + i] = 0.f;
    for (int i = t; i < 16;      i += 256) ws[WS_Z + i] = 0.f;
    for (int i = t; i < B_ * O_; i += 256) ws[WS_OIA + i] = 0.f;
  }
}

// ------------------------------------------- fused GEMM + norm + clip + exp
// D = Qpad[16 x K] x Feats^T tile, one wave per 16 database rows.
__global__ void gemm_exp_kernel(const float* __restrict__ qpad,
                                const float* __restrict__ feats,
                                int Kdim, float* __restrict__ eatt) {
  const int wave = threadIdx.x >> 5;
  const int tile = blockIdx.x * 4 + wave;        // wave-uniform
  if (tile * 16 >= N_) return;                    // uniform scalar branch
  const int lane = threadIdx.x & 31;
  const int col  = lane & 15;
  const int koff = (lane >= 16) ? 2 : 0;          // ISA A/B frag K split

  const float* qrow = qpad + (size_t)col * Kdim + koff;
  const float* frow = feats + (size_t)(tile * 16 + col) * Kdim + koff;

  v8f c = {};
  float sq = 0.f;
  for (int k = 0; k < Kdim; k += 4) {
    v2f a = *(const v2f*)(qrow + k);
    v2f b = *(const v2f*)(frow + k);
    sq += b.x * b.x + b.y * b.y;
    c = __builtin_amdgcn_wmma_f32_16x16x4_f32(false, a, false, b, (short)0, c,
                                              false, false);
  }
  // combine the two K-halves of the per-column sum of squares
  const float tot = sq + __shfl_xor(sq, 16, 32);
  const float inv = 1.0f / fmaxf(sqrtf(tot), 1e-12f);
  if (lane < 16) {                                // rows 0..7 are the 8 batches
    const int n = tile * 16 + col;
#pragma unroll
    for (int r = 0; r < 8; ++r) {
      float s = c[r] * inv;
      s = fminf(fmaxf(s, 0.f), 1.f);
      eatt[(size_t)r * N_ + n] = __expf(s);
    }
  }
}

// ---------------------------------------------------------------- Z sums
__global__ void zsum_kernel(const float* __restrict__ e1,
                            const float* __restrict__ e2,
                            float* __restrict__ Z) {
  __shared__ float red[256];
  const int row = blockIdx.y;  // 0..15
  const float* src = (row < 8) ? (e1 + (size_t)row * N_)
                               : (e2 + (size_t)(row - 8) * N_);
  float s = 0.f;
  for (int n = blockIdx.x * 256 + threadIdx.x; n < N_; n += gridDim.x * 256)
    s += src[n];
  red[threadIdx.x] = s; __syncthreads();
  for (int off = 128; off > 0; off >>= 1) {
    if (threadIdx.x < off) red[threadIdx.x] += red[threadIdx.x + off];
    __syncthreads();
  }
  if (threadIdx.x == 0) atomicAdd(&Z[row], red[0]);
}

// ------------------------------------------------------- gated softmax mix
__global__ void mix_kernel(float* __restrict__ e1, const float* __restrict__ e2,
                           const float* __restrict__ Z,
                           const float* __restrict__ temp) {
  const float aGate = 1.f / (1.f + __expf(-temp[0]));
  const int i = blockIdx.x * 256 + threadIdx.x;
  if (i >= B_ * N_) return;
  const int b = i / N_;
  e1[i] = aGate * e1[i] / Z[b] + (1.f - aGate) * e2[i] / Z[8 + b];
}

// ------------------------------------------- per-row k-th value by bisection
__global__ void threshold_kernel(const float* __restrict__ att,
                                 float* __restrict__ thr) {
  __shared__ float redf[1024];
  __shared__ int   redi[1024];
  __shared__ float bounds[2];
  const int b = blockIdx.x;
  const int t = threadIdx.x;
  const float* row = att + (size_t)b * N_;

  float m = 0.f;
  for (int n = t; n < N_; n += 1024) m = fmaxf(m, row[n]);
  redf[t] = m; __syncthreads();
  for (int off = 512; off > 0; off >>= 1) {
    if (t < off) redf[t] = fmaxf(redf[t], redf[t + off]);
    __syncthreads();
  }
  if (t == 0) { bounds[0] = 0.f; bounds[1] = redf[0] * 1.0000001f + 1e-30f; }
  __syncthreads();

  for (int it = 0; it < 34; ++it) {
    const float mid = 0.5f * (bounds[0] + bounds[1]);
    int c = 0;
    for (int n = t; n < N_; n += 1024) c += (row[n] >= mid) ? 1 : 0;
    redi[t] = c; __syncthreads();
    for (int off = 512; off > 0; off >>= 1) {
      if (t < off) redi[t] += redi[t + off];
      __syncthreads();
    }
    if (t == 0) { if (redi[0] > TOPK_) bounds[0] = mid; else bounds[1] = mid; }
    __syncthreads();
  }
  if (t == 0) thr[b] = bounds[1];
}

// ------------------------------------------- union-over-batch top-k masking
__global__ void mask_kernel(float* __restrict__ att,
                            const float* __restrict__ thr) {
  const int n = blockIdx.x * 256 + threadIdx.x;
  if (n >= N_) return;
  bool keep = false;
#pragma unroll
  for (int b = 0; b < 8; ++b) keep = keep || (att[(size_t)b * N_ + n] >= thr[b]);
  if (!keep) {
#pragma unroll
    for (int b = 0; b < 8; ++b) att[(size_t)b * N_ + n] = 0.f;
  }
}

// ---------------------------------------- oia = att_masked[8xN] @ n_answ[NxO]
#define OIA_CHUNK 1024
__global__ void oia_kernel(const float* __restrict__ att,
                           const float* __restrict__ answ,
                           float* __restrict__ oia) {
  const int wave = threadIdx.x >> 5;   // 0..7, each owns 3 of 24 output tiles
  const int lane = threadIdx.x & 31;
  const int col  = lane & 15;
  const int koff = (lane >= 16) ? 2 : 0;
  const int nStart = blockIdx.x * OIA_CHUNK;
  const int nEnd   = min(nStart + OIA_CHUNK, N_);
  const int m  = col;
  const int o0 = wave * 16, o1 = (wave + 8) * 16, o2 = (wave + 16) * 16;

  v8f acc0 = {}, acc1 = {}, acc2 = {};
  for (int n = nStart; n < nEnd; n += 4) {
    v2f a;
    if (m < 8) a = *(const v2f*)(att + (size_t)m * N_ + n + koff);
    else       { a.x = 0.f; a.y = 0.f; }
    // wave-uniform skip of fully-masked 4-column groups (scalar branch via ballot)
    unsigned long long any = __ballot(a.x != 0.f || a.y != 0.f);
    if (any) {
      const float* r0 = answ + (size_t)(n + koff) * O_;
      const float* r1 = answ + (size_t)(n + koff + 1) * O_;
      v2f b0, b1, b2;
      b0.x = r0[o0 + col]; b0.y = r1[o0 + col];
      b1.x = r0[o1 + col]; b1.y = r1[o1 + col];
      b2.x = r0[o2 + col]; b2.y = r1[o2 + col];
      acc0 = __builtin_amdgcn_wmma_f32_16x16x4_f32(false, a, false, b0, (short)0, acc0, false, false);
      acc1 = __builtin_amdgcn_wmma_f32_16x16x4_f32(false, a, false, b1, (short)0, acc1, false, false);
      acc2 = __builtin_amdgcn_wmma_f32_16x16x4_f32(false, a, false, b2, (short)0, acc2, false, false);
    }
  }
  if (lane < 16) {
#pragma unroll
    for (int r = 0; r < 8; ++r) {
      atomicAdd(&oia[r * O_ + o0 + col], acc0[r]);
      atomicAdd(&oia[r * O_ + o1 + col], acc1[r]);
      atomicAdd(&oia[r * O_ + o2 + col], acc2[r]);
    }
  }
}

// ---------------------------------------------------------------- scores
__global__ void scores_kernel(const float* __restrict__ oia,
                              const float* __restrict__ o,
                              float* __restrict__ out) {
  __shared__ float red[128];
  const int s = blockIdx.x;            // 0..23
  const int b = s / 3, j = s % 3;
  const float* x = oia + (size_t)b * O_;
  const float* y = o + ((size_t)b * 3 + j) * O_;
  float acc = 0.f;
  for (int i = threadIdx.x; i < O_; i += 128) acc += x[i] * y[i];
  red[threadIdx.x] = acc; __syncthreads();
  for (int off = 64; off > 0; off >>= 1) {
    if (threadIdx.x < off) red[threadIdx.x] += red[threadIdx.x + off];
    __syncthreads();
  }
  if (threadIdx.x == 0) out[s] = red[0];
}

// ---------------------------------------------------------------- launch
extern "C" void kernel_launch(void* const* d_in, const int* in_sizes, int n_in,
                              void* d_out, int out_size, void* d_ws, size_t ws_size,
                              hipStream_t stream) {
  (void)in_sizes; (void)n_in; (void)out_size; (void)ws_size;
  const float* v      = (const float*)d_in[0];
  const float* nfeats = (const float*)d_in[1];
  const float* aud    = (const float*)d_in[2];
  const float* nauds  = (const float*)d_in[3];
  const float* o      = (const float*)d_in[4];
  const float* nansw  = (const float*)d_in[5];
  const float* temp   = (const float*)d_in[6];

  float* ws  = (float*)d_ws;
  float* q1  = ws + WS_Q1;
  float* q2  = ws + WS_Q2;
  float* e1  = ws + WS_E1;
  float* e2  = ws + WS_E2;
  float* Z   = ws + WS_Z;
  float* thr = ws + WS_THR;
  float* oia = ws + WS_OIA;

  prep_kernel<<<17, 256, 0, stream>>>(v, aud, ws);

  const int ntiles = N_ / 16;  // 3125
  gemm_exp_kernel<<<(ntiles + 3) / 4, 128, 0, stream>>>(q1, nfeats, V_, e1);
  gemm_exp_kernel<<<(ntiles + 3) / 4, 128, 0, stream>>>(q2, nauds,  A_, e2);

  zsum_kernel<<<dim3(32, 16), 256, 0, stream>>>(e1, e2, Z);
  mix_kernel<<<(B_ * N_ + 255) / 256, 256, 0, stream>>>(e1, e2, Z, temp);
  threshold_kernel<<<8, 1024, 0, stream>>>(e1, thr);
  mask_kernel<<<(N_ + 255) / 256, 256, 0, stream>>>(e1, thr);
  oia_kernel<<<(N_ + OIA_CHUNK - 1) / OIA_CHUNK, 256, 0, stream>>>(e1, nansw, oia);
  scores_kernel<<<24, 128, 0, stream>>>(oia, o, (float*)d_out);
}